// PiecewiseLinearActivation_29334626632057
// MI455X (gfx1250) — compile-verified
//
#include <hip/hip_runtime.h>
#include <hip/hip_bf16.h>

// Piecewise-linear activation: out = sum_s slope_s * max(x - t_s, 0)
// Closed form: with c = #{s : t_s <= x},  out = A_c * x - B_c, where
//   A_c = prefix-sum of slopes, B_c = prefix-sum of slope_s * t_s.
// Memory-bound (1 GiB traffic @ 23.3 TB/s ~= 46 us). Strategy: precompute the
// 129-entry (A,B) LUT once, then stream x with vectorized non-temporal B128
// accesses, per-lane LDS lookup (ds_load_b64), one FMA per element.

typedef float v4f __attribute__((ext_vector_type(4)));
typedef float v2f __attribute__((ext_vector_type(2)));

#define NSEG_MAX 128

// ---------------------------------------------------------------------------
// Kernel 1: build LUT in workspace.
// ws layout (floats): [0]=grid0, [1]=invStep, [2..] = 129 interleaved (A,B)
// ---------------------------------------------------------------------------
__global__ void pwl_build_lut(const float* __restrict__ slopes,
                              const float* __restrict__ grid,
                              float* __restrict__ ws, int nseg) {
    if (threadIdx.x == 0 && blockIdx.x == 0) {
        float g0   = grid[0];
        float step = grid[1] - grid[0];
        ws[0] = g0;
        ws[1] = (step != 0.0f) ? (1.0f / step) : 0.0f;
        float* lut = ws + 2;           // interleaved A,B pairs
        float A = 0.0f, B = 0.0f;
        lut[0] = 0.0f;
        lut[1] = 0.0f;
        for (int s = 0; s < nseg; ++s) {
            float sl = slopes[s];
            A += sl;
            B += sl * grid[s];
            lut[2 * (s + 1) + 0] = A;
            lut[2 * (s + 1) + 1] = B;
        }
    }
}

// ---------------------------------------------------------------------------
// Kernel 2: streaming evaluation. 256 threads = 8 wave32s per block.
// ---------------------------------------------------------------------------
__global__ __launch_bounds__(256) void pwl_eval(
        const v4f* __restrict__ x4, const float* __restrict__ ws,
        v4f* __restrict__ out4, long long n4, long long n, float segMaxF) {
    __shared__ v2f s_lut[NSEG_MAX + 1];

    // Stage (A,B) LUT into LDS (<= 1032 bytes).
    const v2f* glut = (const v2f*)(ws + 2);
    int nentries = (int)segMaxF + 1;
    for (int i = threadIdx.x; i < nentries; i += blockDim.x) {
        s_lut[i] = glut[i];
    }
    const float g0      = ws[0];
    const float invStep = ws[1];
    __syncthreads();

    const long long gid     = (long long)blockIdx.x * blockDim.x + threadIdx.x;
    const long long gstride = (long long)gridDim.x * blockDim.x;

    for (long long i = gid; i < n4; i += gstride) {
        v4f v = __builtin_nontemporal_load(&x4[i]);   // global_load_b128 NT
        v4f r;
#pragma unroll
        for (int k = 0; k < 4; ++k) {
            float xv = v[k];
            // c = clamp(floor((x-g0)/step)+1, 0, nseg); ReLU term is 0 at the
            // boundary, so ulp-level rounding of the index is value-neutral.
            float t = __builtin_floorf((xv - g0) * invStep) + 1.0f;
            t = fminf(fmaxf(t, 0.0f), segMaxF);
            int c = (int)t;
            v2f ab = s_lut[c];                        // ds_load_b64
            r[k] = __builtin_fmaf(ab[0], xv, -ab[1]); // A*x - B
        }
        __builtin_nontemporal_store(r, &out4[i]);     // global_store_b128 NT
    }

    // Scalar tail for n not divisible by 4 (not hit for 8*4096*4096).
    const float* x   = (const float*)x4;
    float*       out = (float*)out4;
    for (long long i = n4 * 4 + gid; i < n; i += gstride) {
        float xv = x[i];
        float t = fminf(fmaxf(__builtin_floorf((xv - g0) * invStep) + 1.0f,
                              0.0f), segMaxF);
        v2f ab = s_lut[(int)t];
        out[i] = __builtin_fmaf(ab[0], xv, -ab[1]);
    }
}

// ---------------------------------------------------------------------------
// Launch
// ---------------------------------------------------------------------------
extern "C" void kernel_launch(void* const* d_in, const int* in_sizes, int n_in,
                              void* d_out, int out_size, void* d_ws, size_t ws_size,
                              hipStream_t stream) {
    const float* x      = (const float*)d_in[0];
    const float* slopes = (const float*)d_in[1];
    const float* grid   = (const float*)d_in[2];
    float*       out    = (float*)d_out;
    float*       ws     = (float*)d_ws;

    const long long n    = (long long)in_sizes[0];
    int             nseg = in_sizes[1];
    if (nseg > NSEG_MAX) nseg = NSEG_MAX;

    // Build the prefix LUT (runs in ~128 serial iterations, once per call).
    pwl_build_lut<<<1, 1, 0, stream>>>(slopes, grid, ws, nseg);

    // Main streaming pass: ~4 float4 iterations per thread to amortize the
    // per-block LDS LUT fill while keeping plenty of blocks for all WGPs.
    const long long n4 = n / 4;
    const int threads = 256;
    long long blocksLL = (n4 + (long long)threads * 4 - 1) / ((long long)threads * 4);
    if (blocksLL < 1) blocksLL = 1;
    if (blocksLL > 0x7FFFFFFFLL) blocksLL = 0x7FFFFFFFLL;
    const int blocks = (int)blocksLL;

    pwl_eval<<<blocks, threads, 0, stream>>>(
        (const v4f*)x, ws, (v4f*)out, n4, n, (float)nseg);
}